// PCRenderer_77489799954476
// MI455X (gfx1250) — compile-verified
//
#include <hip/hip_runtime.h>

// Point-cloud splat renderer for MI455X (gfx1250).
// Kernel 1: projection via V_WMMA_F32_16X16X4_F32 (camera rows in A, 16 homogeneous
//           points in B; D lane n holds Xc of point n in d[0..2]).
// Kernel 2: per-16x16-pixel-tile gather; the packed (u,v,z,pad) table is streamed
//           through LDS with double-buffered global_load_async_to_lds_b128 +
//           s_wait_asynccnt (CDNA5 async-copy path), top-5-by-z kept in registers.

#define IMG  512
#define KPP  5
#define TILE 16
#define RADIUS 0.007f
#define CHUNKSZ 256

typedef __attribute__((ext_vector_type(2))) float v2f;
typedef __attribute__((ext_vector_type(8))) float v8f;

#if __has_builtin(__builtin_amdgcn_wmma_f32_16x16x4_f32)
#define HAS_WMMA4 1
#endif

__global__ __launch_bounds__(256) void pc_project(
    const float* __restrict__ verts,
    const float* __restrict__ camR,
    const float* __restrict__ camT,
    const float* __restrict__ camK,
    float4* __restrict__ uvzw,
    int P)
{
  const float K00 = camK[0], K02 = camK[2], K11 = camK[4], K12 = camK[5];
#ifdef HAS_WMMA4
  const int gid  = blockIdx.x * blockDim.x + threadIdx.x;
  const int wave = gid >> 5;          // one 16-point group per wave32
  const int lane = gid & 31;
  const int m    = lane & 15;
  const int hi   = lane >> 4;
  const int base = wave * 16;

  // A (16x4, MxK), f32 layout: lanes 0-15 carry K=0,1 ; lanes 16-31 carry K=2,3.
  // Row m (m<3) = [R[m][0], R[m][1], R[m][2], t[m]]; rows 3..15 = 0.
  v2f a; a.x = 0.0f; a.y = 0.0f;
  if (m < 3) {
    if (hi == 0) { a.x = camR[m*3+0]; a.y = camR[m*3+1]; }
    else         { a.x = camR[m*3+2]; a.y = camT[m];     }
  }
  // B (4x16, KxN): column n = [x, y, z, 1] of point base+n.
  // Lanes 0-15: K=0,1 of column (lane); lanes 16-31: K=2,3 of column (lane-16).
  const int p  = base + m;
  const float vx = verts[3*p+0], vy = verts[3*p+1], vz = verts[3*p+2];
  v2f b; b.x = hi ? vz : vx; b.y = hi ? 1.0f : vy;

  v8f c = {};
  // D = A*B : D[m][n] = Xc_m of point n
  v8f d = __builtin_amdgcn_wmma_f32_16x16x4_f32(
      /*neg_a=*/false, a, /*neg_b=*/false, b,
      /*c_mod=*/(short)0, c, /*reuse_a=*/false, /*reuse_b=*/false);

  if (hi == 0) {  // lane n<16 holds D[0..7][n]: Xc.x, Xc.y, Xc.z in d[0..2]
    const float X0 = d[0], X1 = d[1], zc = d[2];
    const float u = K00 * X0 / zc + K02;
    const float v = K11 * X1 / zc + K12;
    uvzw[p] = make_float4(u, v, zc, 0.0f);
  }
#else
  const int p = blockIdx.x * blockDim.x + threadIdx.x;  // 2x threads launched; guard
  if (p < P) {
    const float vx = verts[3*p], vy = verts[3*p+1], vz = verts[3*p+2];
    const float X0 = camR[0]*vx + camR[1]*vy + camR[2]*vz + camT[0];
    const float X1 = camR[3]*vx + camR[4]*vy + camR[5]*vz + camT[1];
    const float zc = camR[6]*vx + camR[7]*vy + camR[8]*vz + camT[2];
    uvzw[p] = make_float4(K00*X0/zc + K02, K11*X1/zc + K12, zc, 0.0f);
  }
#endif
}

__global__ __launch_bounds__(256) void pc_render(
    const float4* __restrict__ uvzw,
    const float*  __restrict__ colors,
    float* __restrict__ out,
    int P)
{
  __shared__ float4 buf[2][CHUNKSZ];

  const int t  = threadIdx.y * TILE + threadIdx.x;
  const int px = blockIdx.x * TILE + threadIdx.x;
  const int py = blockIdx.y * TILE + threadIdx.y;
  const float pxc   = (float)px + 0.5f;
  const float pyc   = (float)py + 0.5f;
  const float scale = 2.0f / (float)IMG;
  const float r2    = RADIUS * RADIUS;

  float zt[KPP], d2t[KPP];
  int   it[KPP];
#pragma unroll
  for (int k = 0; k < KPP; ++k) { zt[k] = __builtin_inff(); d2t[k] = 0.0f; it[k] = -1; }

  // LDS byte offsets for this thread's staging slot (generic addr low 32 bits ==
  // wave-relative LDS offset per CDNA5 aperture mapping).
  const unsigned lds0 = (unsigned)(unsigned long long)(const void*)&buf[0][t];
  const unsigned lds1 = (unsigned)(unsigned long long)(const void*)&buf[1][t];

  // Prologue: chunk 0 -> buf0 (one b128 per lane; 256 threads cover the chunk).
  asm volatile("global_load_async_to_lds_b128 %0, %1, off"
               :: "v"(lds0), "v"((unsigned long long)(uvzw + t)) : "memory");

  const int nchunk = P / CHUNKSZ;
  for (int c = 0; c < nchunk; ++c) {
    if (c + 1 < nchunk) {
      const unsigned dst = ((c + 1) & 1) ? lds1 : lds0;
      asm volatile("global_load_async_to_lds_b128 %0, %1, off"
                   :: "v"(dst),
                      "v"((unsigned long long)(uvzw + (c + 1) * CHUNKSZ + t))
                   : "memory");
      asm volatile("s_wait_asynccnt 0x1" ::: "memory");  // my chunk-c lanes done
    } else {
      asm volatile("s_wait_asynccnt 0x0" ::: "memory");
    }
    __syncthreads();  // all waves' chunk-c data visible

    const float4* cur = buf[c & 1];
#pragma unroll 4
    for (int j = 0; j < CHUNKSZ; ++j) {
      const float4 q = cur[j];                 // LDS broadcast read
      const float dx = (q.x - pxc) * scale;
      const float dy = (q.y - pyc) * scale;
      const float d2 = dx * dx + dy * dy;
      if (d2 < r2 && q.z > 0.0f && q.z < zt[KPP - 1]) {
        // strict-< bubble insert: keeps ascending z, earlier index wins ties
        float cz = q.z, cd = d2;
        int   ci = c * CHUNKSZ + j;
#pragma unroll
        for (int k = 0; k < KPP; ++k) {
          if (cz < zt[k]) {
            const float tz = zt[k], td = d2t[k]; const int ti = it[k];
            zt[k] = cz; d2t[k] = cd; it[k] = ci;
            cz = tz; cd = td; ci = ti;
          }
        }
      }
    }
    __syncthreads();  // done reading before next overwrite of this buffer
  }

  // Front-to-back alpha compositing (exclusive transmittance).
  float T = 1.0f, r = 0.0f, g = 0.0f, b = 0.0f;
#pragma unroll
  for (int k = 0; k < KPP; ++k) {
    const bool  valid = it[k] >= 0;
    const int   ic    = valid ? it[k] : 0;
    const float alpha = valid ? (1.0f - d2t[k] / r2) : 0.0f;
    const float w     = alpha * T;
    r += w * colors[3 * ic + 0];
    g += w * colors[3 * ic + 1];
    b += w * colors[3 * ic + 2];
    T *= (1.0f - alpha);
  }
  if (it[0] < 0) { r = 1.0f; g = 1.0f; b = 1.0f; }  // white background

  const int o = (py * IMG + px) * 3;
  out[o + 0] = r; out[o + 1] = g; out[o + 2] = b;
}

extern "C" void kernel_launch(void* const* d_in, const int* in_sizes, int n_in,
                              void* d_out, int out_size, void* d_ws, size_t ws_size,
                              hipStream_t stream) {
  const float* verts = (const float*)d_in[0];  // (1,P,3)
  const float* cols  = (const float*)d_in[1];  // (1,P,3)
  const float* camR  = (const float*)d_in[2];  // (3,3)
  const float* camT  = (const float*)d_in[3];  // (3,)
  const float* camK  = (const float*)d_in[4];  // (3,3)
  float*       out   = (float*)d_out;          // (512,512,3)

  const int P = in_sizes[0] / 3;               // 16384
  float4* uvzw = (float4*)d_ws;                // P * 16 bytes of scratch

  // 2 lanes per point in the WMMA path -> P/128 blocks of 256 covers P/16 waves.
  pc_project<<<dim3(P / 128), dim3(256), 0, stream>>>(verts, camR, camT, camK, uvzw, P);

  dim3 block(TILE, TILE);
  dim3 grid(IMG / TILE, IMG / TILE);
  pc_render<<<grid, block, 0, stream>>>(uvzw, cols, out, P);
}